// ModelNew_80908593922842
// MI455X (gfx1250) — compile-verified
//
#include <hip/hip_runtime.h>
#include <math.h>

typedef __attribute__((ext_vector_type(2))) float v2f;
typedef __attribute__((ext_vector_type(8))) float v8f;
typedef __attribute__((ext_vector_type(4))) unsigned int u32x4;
typedef __attribute__((ext_vector_type(4))) int i32x4;
typedef __attribute__((ext_vector_type(8))) int i32x8;

// Problem constants (from reference)
#define BATCH   64
#define CIN     3
#define DDIM    16
#define HDIM    64
#define WDIM    64
#define COUT    24
#define DOUT    14
#define HOUT    62
#define WOUT    62
#define PTOT    (HOUT * WOUT)        // 3844 output positions per (b, d')
#define KRED    81                   // CIN * 3*3*3 reduction length
#define KPAD    84
#define KSTEPS  21                   // KPAD / 4
#define NTILES  2                    // 16-position N tiles per wave
#define WAVES_PER_BLK 8
#define POS_PER_BLK   256            // 8 waves * 32 positions
#define BLOCKS_X ((PTOT + POS_PER_BLK - 1) / POS_PER_BLK)   // 16

// x slab in LDS: [cin][kd][8 h-rows][64 w] floats, double buffered
#define SLAB_ROWS 8
#define SLAB_ELEMS (CIN * 3 * SLAB_ROWS * WDIM)   // 4608 floats = 18 KB

// Offset within the LDS slab for reduction index k = ci*27 + kd*9 + kh*3 + kw
// slab index = (ci*3 + kd)*512 + (kh + h_local)*64 + (kw + w)  -> constant part:
__host__ __device__ constexpr int soff(int k) {
    int ci = k / 27;
    int r  = k % 27;
    int kd = r / 9;
    int r2 = r % 9;
    int kh = r2 / 3;
    int kw = r2 % 3;
    return (ci * 3 + kd) * (SLAB_ROWS * WDIM) + kh * WDIM + kw;
}

__global__ __launch_bounds__(256) void conv3d_min_softmax_wmma(
    const float* __restrict__ x,      // [64,3,16,64,64]
    const float* __restrict__ w,      // [24,81]
    const float* __restrict__ bias,   // [24]
    float* __restrict__ out)          // [64,24,62,62]
{
    __shared__ __align__(16) float wlds[32 * KPAD];          // 10.5 KB weights
    __shared__ __align__(16) float slab[2 * SLAB_ELEMS];     // 2 x 18 KB x tiles

    // Stage weights in LDS, zero-padded to 32 rows x 84 K
    for (int idx = threadIdx.x; idx < 32 * KPAD; idx += 256) {
        int row = idx / KPAD;
        int k   = idx - row * KPAD;
        float v = 0.0f;
        if (row < COUT && k < KRED) v = w[row * KRED + k];
        wlds[idx] = v;
    }

    const int lane = threadIdx.x & 31;
    const int wave = threadIdx.x >> 5;
    const int b    = blockIdx.y;                 // block-uniform batch
    const int P0   = blockIdx.x * POS_PER_BLK;   // block's first output position
    const int h0   = P0 / WOUT;                  // first h' row touched by block
    const int p0   = P0 + wave * (16 * NTILES);

    const int n  = lane & 15;
    const int hi = lane >> 4;

    // Per-tile positions and slab-local spatial offsets (clamped for OOB lanes)
    int p[NTILES], nloc[NTILES];
#pragma unroll
    for (int s = 0; s < NTILES; ++s) {
        p[s] = p0 + 16 * s + n;
        int pcl = (p[s] < PTOT) ? p[s] : (PTOT - 1);
        int ho  = pcl / WOUT;
        int wo  = pcl - ho * WOUT;
        nloc[s] = (ho - h0) * WDIM + wo;          // 0 <= ho-h0 <= 5
    }
    const int delta = nloc[1] - nloc[0];

    // Precomputed per-lane LDS element offsets (slab-relative, depth-invariant).
    // B fragment 4x16: element0 k = 4j+hi, element1 k = 4j+2+hi.
    int offA[KSTEPS], offB[KSTEPS];
#pragma unroll
    for (int j = 0; j < KSTEPS; ++j) {
        int c0 = (4 * j + 1 < KRED) ? (hi ? soff(4 * j + 1) : soff(4 * j))
                                    : soff(80);   // j==20: hi==1 zero-filled at use
        int c1 = (4 * j + 3 < KRED) ? (hi ? soff(4 * j + 3) : soff(4 * j + 2))
                                    : 0;          // j==20: padded, never loaded
        offA[j] = c0 + nloc[0];
        offB[j] = c1 + nloc[0];
    }

    const int arowLo = (lane & 15) * KPAD;
    const int arowHi = ((lane & 15) + 16) * KPAD;

    // --- TDM setup (uniform, wave 0 issues) ---
    const bool issuer = (__builtin_amdgcn_readfirstlane(wave) == 0);
    const float* xbh = x + (size_t)b * (CIN * DDIM * HDIM * WDIM) + (size_t)h0 * WDIM;
    const unsigned slab_lds0 = (unsigned)(size_t)(void*)&slab[0];
    const unsigned slab_lds1 = (unsigned)(size_t)(void*)&slab[SLAB_ELEMS];

    // Descriptor group1/2/3 parts that don't change (strides in data_size units):
    // tile = 64(w) x 8(h) x 3(kd) x 3(ci); strides 64 / 4096 / 65536.
    i32x8 g1;
    g1[0] = 0x00020000;                    // workgroup_mask=0, data_size=4B
    g1[1] = (64 << 16);                    // tensor_dim0 = 64
    g1[2] = ((64 - h0) & 0xFFFF) << 16;    // tensor_dim1 = remaining h rows
    g1[3] = (64 << 16);                    // tile_dim0 = 64
    g1[4] = SLAB_ROWS | (3 << 16);         // tile_dim1 = 8, tile_dim2 = 3
    g1[5] = 64;                            // tensor_dim0_stride = 64
    g1[6] = (4096 << 16);                  // tensor_dim1_stride = 4096
    g1[7] = 0;
    i32x4 g3 = {0, 0, 0, 0};
    i32x8 g5 = {0, 0, 0, 0, 0, 0, 0, 0};   // clang-23 6-arg form: extra group, zeros

    // Issue first slab (d = 0) before the depth loop.
    __syncthreads();   // weights staged; slab bufs free
    if (issuer) {
        unsigned long long ga = (unsigned long long)(size_t)(const void*)xbh;
        u32x4 g0;
        g0[0] = 1u;                                        // count=1, user mode
        g0[1] = slab_lds0;
        g0[2] = (unsigned)ga;
        g0[3] = (unsigned)((ga >> 32) & 0x01FFFFFF) | 0x80000000u;  // type=2
        i32x4 g2;
        g2[0] = DDIM;                                      // tensor_dim2 (d slices)
        g2[1] = CIN;                                       // tensor_dim3
        g2[2] = 65536;                                     // tensor_dim2_stride
        g2[3] = (CIN << 16);                               // tile_dim3 = 3
        __builtin_amdgcn_tensor_load_to_lds(g0, g1, g2, g3, g5, 0);
    }

    v8f mnLo[NTILES], mnHi[NTILES];
#pragma unroll
    for (int s = 0; s < NTILES; ++s)
#pragma unroll
        for (int r = 0; r < 8; ++r) {
            mnLo[s][r] = __builtin_inff();
            mnHi[s][r] = __builtin_inff();
        }

#pragma unroll 1
    for (int d = 0; d < DOUT; ++d) {
        // Prefetch next depth slab into the other buffer (double buffering).
        if (issuer && (d + 1 < DOUT)) {
            unsigned long long ga =
                (unsigned long long)(size_t)(const void*)(xbh + (size_t)(d + 1) * (HDIM * WDIM));
            u32x4 g0;
            g0[0] = 1u;
            g0[1] = ((d + 1) & 1) ? slab_lds1 : slab_lds0;
            g0[2] = (unsigned)ga;
            g0[3] = (unsigned)((ga >> 32) & 0x01FFFFFF) | 0x80000000u;
            i32x4 g2;
            g2[0] = DDIM - (d + 1);    // remaining d slices (edge -> zeros)
            g2[1] = CIN;
            g2[2] = 65536;             // tensor_dim2_stride
            g2[3] = (CIN << 16);       // tile_dim3 = 3
            __builtin_amdgcn_tensor_load_to_lds(g0, g1, g2, g3, g5, 0);
            __builtin_amdgcn_s_wait_tensorcnt(1);   // slab d complete
        } else if (issuer) {
            __builtin_amdgcn_s_wait_tensorcnt(0);
        }
        __syncthreads();               // slab d visible to all waves

        const float* sd = &slab[(d & 1) ? SLAB_ELEMS : 0];
        v8f acc0a = {}, acc1a = {}, acc0b = {}, acc1b = {};
#pragma unroll
        for (int j = 0; j < KSTEPS; ++j) {
            // A fragments (16x4 f32 weights) from LDS
            const int kb = 4 * j + 2 * hi;
            v2f aLo = *(const v2f*)&wlds[arowLo + kb];
            v2f aHi = *(const v2f*)&wlds[arowHi + kb];

            // B fragments from the slab (pure LDS, zero address math)
            float l0 = sd[offA[j]];
            float m0 = sd[offA[j] + delta];
            v2f bf0, bf1;
            if (4 * j + 1 < KRED) { bf0[0] = l0; bf1[0] = m0; }
            else                  { bf0[0] = hi ? 0.0f : l0;
                                    bf1[0] = hi ? 0.0f : m0; }
            if (4 * j + 3 < KRED) { bf0[1] = sd[offB[j]];
                                    bf1[1] = sd[offB[j] + delta]; }
            else                  { bf0[1] = 0.0f; bf1[1] = 0.0f; }

            acc0a = __builtin_amdgcn_wmma_f32_16x16x4_f32(
                false, aLo, false, bf0, (short)0, acc0a, false, false);
            acc1a = __builtin_amdgcn_wmma_f32_16x16x4_f32(
                false, aHi, false, bf0, (short)0, acc1a, false, false);
            acc0b = __builtin_amdgcn_wmma_f32_16x16x4_f32(
                false, aLo, false, bf1, (short)0, acc0b, false, false);
            acc1b = __builtin_amdgcn_wmma_f32_16x16x4_f32(
                false, aHi, false, bf1, (short)0, acc1b, false, false);
        }
#pragma unroll
        for (int r = 0; r < 8; ++r) {
            mnLo[0][r] = fminf(mnLo[0][r], acc0a[r]);
            mnHi[0][r] = fminf(mnHi[0][r], acc1a[r]);
            mnLo[1][r] = fminf(mnLo[1][r], acc0b[r]);
            mnHi[1][r] = fminf(mnHi[1][r], acc1b[r]);
        }
        __syncthreads();               // reads of slab d done before it is reused
    }

    // ---- bias + softmax over 24 channels, per N tile ----
#pragma unroll
    for (int ts = 0; ts < NTILES; ++ts) {
        float vLo[8], vHi[8];
#pragma unroll
        for (int r = 0; r < 8; ++r) {
            vLo[r] = mnLo[ts][r] + bias[8 * hi + r];
            vHi[r] = mnHi[ts][r] + ((hi == 0) ? bias[16 + r] : 0.0f);
        }

        float mx = -__builtin_inff();
#pragma unroll
        for (int r = 0; r < 8; ++r) mx = fmaxf(mx, vLo[r]);
#pragma unroll
        for (int r = 0; r < 8; ++r)
            mx = fmaxf(mx, (hi == 0) ? vHi[r] : -__builtin_inff());
        mx = fmaxf(mx, __shfl_xor(mx, 16, 32));

        float sum = 0.0f;
        float eLo[8], eHi[8];
#pragma unroll
        for (int r = 0; r < 8; ++r) { eLo[r] = __expf(vLo[r] - mx); sum += eLo[r]; }
#pragma unroll
        for (int r = 0; r < 8; ++r) {
            float e = __expf(vHi[r] - mx);
            eHi[r]  = (hi == 0) ? e : 0.0f;
            sum += eHi[r];
        }
        sum += __shfl_xor(sum, 16, 32);
        const float inv = 1.0f / sum;

        if (p[ts] < PTOT) {
            size_t ob = (size_t)b * COUT * PTOT + p[ts];
#pragma unroll
            for (int r = 0; r < 8; ++r)
                out[ob + (size_t)(8 * hi + r) * PTOT] = eLo[r] * inv;
            if (hi == 0) {
#pragma unroll
                for (int r = 0; r < 8; ++r)
                    out[ob + (size_t)(16 + r) * PTOT] = eHi[r] * inv;
            }
        }
    }
}

extern "C" void kernel_launch(void* const* d_in, const int* in_sizes, int n_in,
                              void* d_out, int out_size, void* d_ws, size_t ws_size,
                              hipStream_t stream) {
    (void)in_sizes; (void)n_in; (void)out_size; (void)d_ws; (void)ws_size;
    const float* x    = (const float*)d_in[0];
    const float* w    = (const float*)d_in[1];
    const float* bias = (const float*)d_in[2];
    float* out        = (float*)d_out;

    dim3 grid(BLOCKS_X, BATCH);   // (16, 64)
    conv3d_min_softmax_wmma<<<grid, 256, 0, stream>>>(x, w, bias, out);
}